// MedMamba_19937238188359
// MI455X (gfx1250) — compile-verified
//
#include <hip/hip_runtime.h>
#include <hip/hip_bf16.h>
#include <math.h>

// Problem constants (from reference)
#define B_  16
#define T_  512
#define C_  16
#define E_  256
#define DI_ 512
#define N_  16
#define R_  16
#define K_  4
#define DBL_ (R_ + 2 * N_)   // 48
#define CHUNK_ 64            // timesteps of B-rows staged per LDS chunk in the scan

typedef __attribute__((ext_vector_type(2))) float        v2f;
typedef __attribute__((ext_vector_type(8))) float        v8f;
typedef __attribute__((ext_vector_type(4))) unsigned int v4u;
typedef __attribute__((ext_vector_type(8))) int          v8i;
typedef __attribute__((ext_vector_type(4))) int          v4i;

// ---- CDNA5 async / TDM feature probes (compile-safe fallbacks) -------------
#if defined(__has_builtin)
# if __has_builtin(__builtin_amdgcn_global_load_async_to_lds_b128) && \
     __has_builtin(__builtin_amdgcn_s_wait_asynccnt)
#  define HAVE_ASYNC_ 1
# endif
# if __has_builtin(__builtin_amdgcn_tensor_load_to_lds) && \
     __has_builtin(__builtin_amdgcn_s_wait_tensorcnt)
#  define HAVE_TDM_ 1
# endif
#endif
#ifndef HAVE_ASYNC_
# define HAVE_ASYNC_ 0
#endif
#ifndef HAVE_TDM_
# define HAVE_TDM_ 0
#endif

#if HAVE_ASYNC_
// Probe-derived signature: (AS(1) int4* src, AS(3) int4* dst, imm offset, imm cpol)
typedef __attribute__((address_space(1))) v4i gv4i;
typedef __attribute__((address_space(3))) v4i lv4i;
__device__ __forceinline__ void async_b128_(const void* g, void* l) {
    __builtin_amdgcn_global_load_async_to_lds_b128(
        (gv4i*)(uintptr_t)g, (lv4i*)(uintptr_t)l, 0, 0);
}
#define WAIT_ASYNC_() __builtin_amdgcn_s_wait_asynccnt(0)
#else
#define WAIT_ASYNC_()
#endif

__device__ __forceinline__ float sigmoidf_(float v) { return 1.0f / (1.0f + __expf(-v)); }
__device__ __forceinline__ float siluf_(float v)    { return v * sigmoidf_(v); }
__device__ __forceinline__ float softplusf_(float v){ return (v > 20.0f) ? v : log1pf(__expf(v)); }

// ---------------------------------------------------------------------------
// Kernel 1: u[b,t,e] = sum_c emb[x[b,t,c], e] + log1p(delta[b,t])*W_time[e] + b_time[e]
// ---------------------------------------------------------------------------
__global__ void k_embed(const int* __restrict__ x, const float* __restrict__ deltas,
                        const float* __restrict__ emb, const float* __restrict__ Wt,
                        const float* __restrict__ bt, float* __restrict__ u)
{
    const int row = blockIdx.x;          // 0 .. B*T-1
    const int e   = threadIdx.x;         // 0 .. 255
    const int* xr = x + row * C_;
    float acc = 0.0f;
#pragma unroll
    for (int c = 0; c < C_; ++c)
        acc += emb[(size_t)xr[c] * E_ + e];
    acc += log1pf(deltas[row]) * Wt[e] + bt[e];
    u[(size_t)row * E_ + e] = acc;
}

// ---------------------------------------------------------------------------
// Kernel 2: fp32 WMMA GEMM:  C[M,N] = A[M,K] @ W[N,K]^T (+bias)(+softplus)
// One wave per 16x16 C tile; V_WMMA_F32_16X16X4_F32, K steps of 4.
// Both A and B fragments are contiguous float2 at row*ld + k0 + 2*half.
// epilogue: 0 = none, 1 = +bias, 2 = softplus(x + bias)
// ---------------------------------------------------------------------------
__global__ void k_gemm_wmma(const float* __restrict__ A, const float* __restrict__ W,
                            const float* __restrict__ bias, float* __restrict__ Cmat,
                            int M, int N, int K, int lda, int ldc, int epilogue)
{
    const int wave    = blockIdx.x * (blockDim.x >> 5) + (threadIdx.x >> 5);
    const int tiles_n = N >> 4;
    const int total   = (M >> 4) * tiles_n;
    if (wave >= total) return;                 // wave-uniform: EXEC stays full
    const int tm   = wave / tiles_n;
    const int tn   = wave - tm * tiles_n;
    const int lane = threadIdx.x & 31;
    const int half = lane >> 4;
    const int l    = lane & 15;

    const float* arow = A + (size_t)(tm * 16 + l) * lda + 2 * half;
    const float* wrow = W + (size_t)(tn * 16 + l) * K   + 2 * half;

    v8f acc = {};
#pragma unroll 4
    for (int k0 = 0; k0 < K; k0 += 4) {
        v2f a = *(const v2f*)(arow + k0);
        v2f b = *(const v2f*)(wrow + k0);
        acc = __builtin_amdgcn_wmma_f32_16x16x4_f32(
                  false, a, false, b, (short)0, acc, false, false);
    }

    const int col = tn * 16 + l;
    const float bv = (epilogue >= 1) ? bias[col] : 0.0f;
#pragma unroll
    for (int r = 0; r < 8; ++r) {
        const int row = tm * 16 + r + 8 * half;
        float v = acc[r] + bv;
        if (epilogue == 2) v = softplusf_(v);
        Cmat[(size_t)row * ldc + col] = v;
    }
}

// ---------------------------------------------------------------------------
// Kernel 3: causal depthwise conv (K=4) + SiLU
// ---------------------------------------------------------------------------
__global__ void k_conv_silu(const float* __restrict__ xz, const float* __restrict__ Wc,
                            const float* __restrict__ bc, float* __restrict__ xc)
{
    const int t = blockIdx.x;
    const int b = blockIdx.y;
    const int d = threadIdx.x;               // 0 .. 511
    const float* base = xz + (size_t)b * T_ * (2 * DI_);
    float acc = bc[d];
#pragma unroll
    for (int k = 0; k < K_; ++k) {
        const int tt = t + k - (K_ - 1);
        if (tt >= 0) acc += base[(size_t)tt * (2 * DI_) + d] * Wc[d * K_ + k];
    }
    xc[((size_t)b * T_ + t) * DI_ + d] = siluf_(acc);
}

// ---------------------------------------------------------------------------
// Kernel 4: selective scan. One block per batch b, 512 threads over d,
// h[16] state in registers. B_t rows staged in CHUNK_-step double-buffered
// LDS chunks; next chunk fetched with GLOBAL_LOAD_ASYNC_TO_LDS_B128
// (ASYNCcnt) while the recurrence runs on the current one. delta/xc rows
// prefetched one step ahead (global_prefetch_b8). Only t = T-1 produces
// output: y = (C_t . h) + xc*D, gated by silu(z_last).
// ---------------------------------------------------------------------------
__global__ void k_scan(const float* __restrict__ dbl, const float* __restrict__ delta,
                       const float* __restrict__ xc, const float* __restrict__ xz,
                       const float* __restrict__ A_log, const float* __restrict__ Dskip,
                       float* __restrict__ ylast)
{
    __shared__ float sB[2][CHUNK_ * N_];     // 2 x 4 KB
    const int b = blockIdx.x;
    const int d = threadIdx.x;

    float a[N_], h[N_];
#pragma unroll
    for (int n = 0; n < N_; ++n) {
        a[n] = -__expf(A_log[d * N_ + n]);
        h[n] = 0.0f;
    }

    // stage one chunk of B rows (16 floats at column R_ of each 48-float row)
    auto stage = [&](int buf, int t0) {
#if HAVE_ASYNC_
        if (d < CHUNK_ * 4) {                // 4 x b128 per row, waves 0-7
            const int r = d >> 2, p = d & 3;
            const float* src = dbl + ((size_t)(b * T_ + t0 + r)) * DBL_ + R_ + p * 4;
            async_b128_(src, &sB[buf][r * N_ + p * 4]);
        }
#else
        for (int i = d; i < CHUNK_ * N_; i += DI_) {
            const int r = i >> 4, n = i & 15;
            sB[buf][i] = dbl[((size_t)(b * T_ + t0 + r)) * DBL_ + R_ + n];
        }
#endif
    };

    int buf = 0;
    stage(buf, 0);
    WAIT_ASYNC_();
    __syncthreads();

    float y_out = 0.0f;
    for (int t0 = 0; t0 < T_; t0 += CHUNK_) {
        if (t0 + CHUNK_ < T_) stage(buf ^ 1, t0 + CHUNK_);   // overlap DMA w/ compute

        for (int tt = 0; tt < CHUNK_; ++tt) {
            const int t = t0 + tt;
            const size_t row = (size_t)b * T_ + t;
            const float del = delta[row * DI_ + d];
            const float xv  = xc[row * DI_ + d];
            if (t + 1 < T_) {
                __builtin_prefetch(&delta[(row + 1) * DI_ + d], 0, 3);
                __builtin_prefetch(&xc[(row + 1) * DI_ + d], 0, 3);
            }
            const float dx = del * xv;
#pragma unroll
            for (int n = 0; n < N_; ++n)
                h[n] = __expf(del * a[n]) * h[n] + dx * sB[buf][tt * N_ + n];

            if (t == T_ - 1) {
                float y = 0.0f;
#pragma unroll
                for (int n = 0; n < N_; ++n)                  // C_t: uniform scalar loads
                    y += h[n] * dbl[row * DBL_ + R_ + N_ + n];
                y += xv * Dskip[d];
                const float z = xz[row * (2 * DI_) + DI_ + d];
                y_out = y * siluf_(z);
            }
        }
        WAIT_ASYNC_();
        __syncthreads();
        buf ^= 1;
    }
    ylast[b * DI_ + d] = y_out;
}

// ---------------------------------------------------------------------------
// Kernel 5: tail. out = LN(ylast @ W_out^T); d_out = [final@W_hc+b, final@W_hr+b]
// ylast (32 KB) pulled into LDS by the Tensor Data Mover (TENSORcnt) when
// available, else cooperative loads. Single block, 256 threads.
// ---------------------------------------------------------------------------
__global__ void k_head(const float* __restrict__ ylast, const float* __restrict__ Wout,
                       const float* __restrict__ ln_g, const float* __restrict__ ln_b,
                       const float* __restrict__ Whc, const float* __restrict__ bhc,
                       const float* __restrict__ Whr, const float* __restrict__ bhr,
                       float* __restrict__ out)
{
    __shared__ float sY[B_ * DI_];   // 32 KB
    __shared__ float sO[B_ * E_];    // 16 KB
    __shared__ float sMu[B_], sRs[B_];
    const int tid = threadIdx.x;     // 0 .. 255

#if HAVE_TDM_
    if (tid < 32) {                  // wave 0 issues one TDM descriptor
        const unsigned long long ga  = (unsigned long long)(uintptr_t)ylast;
        const unsigned int       lds = (unsigned int)(uintptr_t)&sY[0];
        const unsigned int nelem = (unsigned int)(B_ * DI_);       // 8192 x 4B row
        v4u g0;
        g0[0] = 1u;                                         // count=1, user mode
        g0[1] = lds;                                        // lds_addr
        g0[2] = (unsigned int)(ga & 0xFFFFFFFFu);           // global_addr[31:0]
        g0[3] = (unsigned int)((ga >> 32) & 0x01FFFFFFu)    // global_addr[56:32]
              | (2u << 30);                                 // type=2 ("image")
        v8i g1;
        g1[0] = (int)(2u << 16);                            // data_size=4B, mask=0
        g1[1] = (int)((nelem & 0xFFFFu) << 16);             // tensor_dim0[15:0]
        g1[2] = (int)(((nelem >> 16) & 0xFFFFu) | (1u << 16)); // dim0[31:16] | dim1=1
        g1[3] = (int)((nelem & 0xFFFFu) << 16);             // tile_dim0 = nelem
        g1[4] = 1;                                          // tile_dim1 = 1
        g1[5] = (int)nelem;                                 // tensor_dim0_stride[31:0]
        g1[6] = (int)((nelem & 0xFFFFu) << 16);             // stride0 hi | stride1 lo
        g1[7] = 0;
        v4i z4 = {0, 0, 0, 0};
        v8i z8 = {0, 0, 0, 0, 0, 0, 0, 0};
        // 6-arg form on this toolchain: (g0, g1, g2, g3, g4, cpol)
        __builtin_amdgcn_tensor_load_to_lds(g0, g1, z4, z4, z8, 0);
        __builtin_amdgcn_s_wait_tensorcnt(0);
    }
    __syncthreads();
#else
    for (int i = tid; i < B_ * DI_; i += E_) sY[i] = ylast[i];
    __syncthreads();
#endif

    // out[b, e] = sum_k ylast[b,k] * W_out[e,k]  (each thread owns column e)
    float acc[B_];
#pragma unroll
    for (int b = 0; b < B_; ++b) acc[b] = 0.0f;
    for (int k = 0; k < DI_; ++k) {
        const float w = Wout[(size_t)tid * DI_ + k];
#pragma unroll
        for (int b = 0; b < B_; ++b) acc[b] += sY[b * DI_ + k] * w;
    }
#pragma unroll
    for (int b = 0; b < B_; ++b) sO[b * E_ + tid] = acc[b];
    __syncthreads();

    if (tid < B_) {
        float mu = 0.0f;
        for (int e = 0; e < E_; ++e) mu += sO[tid * E_ + e];
        mu *= (1.0f / E_);
        float var = 0.0f;
        for (int e = 0; e < E_; ++e) { float df = sO[tid * E_ + e] - mu; var += df * df; }
        var *= (1.0f / E_);
        sMu[tid] = mu;
        sRs[tid] = rsqrtf(var + 1e-5f);
    }
    __syncthreads();

#pragma unroll
    for (int b = 0; b < B_; ++b)
        sO[b * E_ + tid] = (sO[b * E_ + tid] - sMu[b]) * sRs[b] * ln_g[tid] + ln_b[tid];
    __syncthreads();

    if (tid < B_) {
        float hc = 0.0f, hr = 0.0f;
        for (int e = 0; e < E_; ++e) {
            const float v = sO[tid * E_ + e];
            hc += v * Whc[e];
            hr += v * Whr[e];
        }
        out[tid]      = hc + bhc[0];
        out[B_ + tid] = hr + bhr[0];
    }
}

// ---------------------------------------------------------------------------
extern "C" void kernel_launch(void* const* d_in, const int* in_sizes, int n_in,
                              void* d_out, int out_size, void* d_ws, size_t ws_size,
                              hipStream_t stream)
{
    (void)in_sizes; (void)n_in; (void)out_size; (void)ws_size;
    const int*   x       = (const int*)  d_in[0];
    const float* deltas  = (const float*)d_in[1];
    const float* emb     = (const float*)d_in[2];
    const float* W_time  = (const float*)d_in[3];
    const float* b_time  = (const float*)d_in[4];
    const float* W_in    = (const float*)d_in[5];
    const float* W_conv  = (const float*)d_in[6];
    const float* b_conv  = (const float*)d_in[7];
    const float* W_xproj = (const float*)d_in[8];
    const float* W_dt    = (const float*)d_in[9];
    const float* b_dt    = (const float*)d_in[10];
    const float* A_log   = (const float*)d_in[11];
    const float* D_skip  = (const float*)d_in[12];
    const float* W_out   = (const float*)d_in[13];
    const float* ln_g    = (const float*)d_in[14];
    const float* ln_b    = (const float*)d_in[15];
    const float* W_hc    = (const float*)d_in[16];
    const float* b_hc    = (const float*)d_in[17];
    const float* W_hr    = (const float*)d_in[18];
    const float* b_hr    = (const float*)d_in[19];
    float* out = (float*)d_out;

    // Workspace carve-up (floats): ~77 MB total
    float* ws = (float*)d_ws;
    size_t off = 0;
    float* u     = ws + off; off += (size_t)B_ * T_ * E_;
    float* xz    = ws + off; off += (size_t)B_ * T_ * 2 * DI_;
    float* xc    = ws + off; off += (size_t)B_ * T_ * DI_;
    float* dbl   = ws + off; off += (size_t)B_ * T_ * DBL_;
    float* dl    = ws + off; off += (size_t)B_ * T_ * DI_;
    float* ylast = ws + off; off += (size_t)B_ * DI_;

    // 1) embedding + time embedding
    k_embed<<<B_ * T_, E_, 0, stream>>>(x, deltas, emb, W_time, b_time, u);

    // 2) xz = u @ W_in^T   (M=8192, N=1024, K=256)
    {
        const int M = B_ * T_, N = 2 * DI_, K = E_;
        const int tiles = (M / 16) * (N / 16);
        k_gemm_wmma<<<(tiles + 7) / 8, 256, 0, stream>>>(u, W_in, nullptr, xz,
                                                         M, N, K, E_, N, 0);
    }

    // 3) depthwise causal conv + SiLU
    k_conv_silu<<<dim3(T_, B_), DI_, 0, stream>>>(xz, W_conv, b_conv, xc);

    // 4) dbl = xc @ W_xproj^T  (M=8192, N=48, K=512)
    {
        const int M = B_ * T_, N = DBL_, K = DI_;
        const int tiles = (M / 16) * (N / 16);
        k_gemm_wmma<<<(tiles + 7) / 8, 256, 0, stream>>>(xc, W_xproj, nullptr, dbl,
                                                         M, N, K, DI_, DBL_, 0);
    }

    // 5) delta = softplus(dt @ W_dt^T + b_dt)  (dt = dbl[:, :16] via lda=48; K=16)
    {
        const int M = B_ * T_, N = DI_, K = R_;
        const int tiles = (M / 16) * (N / 16);
        k_gemm_wmma<<<(tiles + 7) / 8, 256, 0, stream>>>(dbl, W_dt, b_dt, dl,
                                                         M, N, K, DBL_, DI_, 2);
    }

    // 6) selective scan over T (sequential), output only at t = T-1
    k_scan<<<B_, DI_, 0, stream>>>(dbl, dl, xc, xz, A_log, D_skip, ylast);

    // 7) W_out GEMM + LayerNorm + two heads
    k_head<<<1, E_, 0, stream>>>(ylast, W_out, ln_g, ln_b, W_hc, b_hc, W_hr, b_hr, out);
}